// BBoxTargetExpand_5291399709104
// MI455X (gfx1250) — compile-verified
//
#include <hip/hip_runtime.h>

// BBoxTargetExpand: the reference's masked select picks the same tensor in
// both branches, so the kernel is a pure streaming copy:
//   d_out[0        .. M*N)  = bbox_targets
//   d_out[M*N .. 2*M*N)     = bbox_weights
// labels is dead and is never read.
//
// Bandwidth-bound: 512 MB total traffic, ~22 us at 23.3 TB/s. Use 128-bit
// vector memory ops with CDNA5 non-temporal hints (read-once / write-once
// streams larger than the 192 MB L2).

typedef float float4_t __attribute__((ext_vector_type(4)));

__global__ __launch_bounds__(256) void bbox_expand_copy_kernel(
    const float4_t* __restrict__ tgt4,
    const float4_t* __restrict__ wgt4,
    float4_t* __restrict__ out_t4,
    float4_t* __restrict__ out_w4,
    const float* __restrict__ tgt_s,
    const float* __restrict__ wgt_s,
    float* __restrict__ out_t_s,
    float* __restrict__ out_w_s,
    long long n4,     // number of float4 elements per tensor
    int rem)          // scalar tail elements per tensor (0..3)
{
    long long i = (long long)blockIdx.x * blockDim.x + threadIdx.x;
    if (i < n4) {
        // global_load_b128 th:TH_LOAD_NT / global_store_b128 th:TH_STORE_NT
        float4_t a = __builtin_nontemporal_load(&tgt4[i]);
        float4_t b = __builtin_nontemporal_load(&wgt4[i]);
        __builtin_nontemporal_store(a, &out_t4[i]);
        __builtin_nontemporal_store(b, &out_w4[i]);
    }
    // Scalar tail (only if total element count not divisible by 4; dead for
    // the reference shape M*N = 32,000,000 but keeps the kernel general).
    if (rem != 0 && blockIdx.x == 0 && (int)threadIdx.x < rem) {
        long long base = n4 * 4 + threadIdx.x;
        out_t_s[base] = tgt_s[base];
        out_w_s[base] = wgt_s[base];
    }
}

extern "C" void kernel_launch(void* const* d_in, const int* in_sizes, int n_in,
                              void* d_out, int out_size, void* d_ws, size_t ws_size,
                              hipStream_t stream) {
    (void)n_in; (void)d_ws; (void)ws_size; (void)out_size;

    const float* tgt = (const float*)d_in[0];   // bbox_targets, M*N f32
    const float* wgt = (const float*)d_in[1];   // bbox_weights, M*N f32
    // d_in[2] = labels (int64) is dead: mask selects identical values.

    const long long nelem = (long long)in_sizes[0];  // M*N
    const long long n4    = nelem >> 2;
    const int       rem   = (int)(nelem & 3LL);

    float* out_t = (float*)d_out;          // first output tensor
    float* out_w = out_t + nelem;          // second output tensor

    const int threads = 256;
    long long work = n4 > 0 ? n4 : 1;      // ensure tail-only case still launches
    unsigned int blocks = (unsigned int)((work + threads - 1) / threads);

    bbox_expand_copy_kernel<<<dim3(blocks), dim3(threads), 0, stream>>>(
        (const float4_t*)tgt, (const float4_t*)wgt,
        (float4_t*)out_t, (float4_t*)out_w,
        tgt, wgt, out_t, out_w,
        n4, rem);
}